// _SelfAttention_31104153157686
// MI455X (gfx1250) — compile-verified
//
#include <hip/hip_runtime.h>
#include <math.h>

// ---- problem dims (match reference) ----
#define BATCH   16
#define HID     2048
#define NH      16
#define NKV     8
#define HD      128
#define BLK     16
#define MAXB    256
#define Q_SIZE  (NH * HD)        // 2048
#define KV_SIZE (NKV * HD)       // 1024
#define QKV_N   (Q_SIZE + 2 * KV_SIZE)   // 4096
#define LMAX    (MAXB * BLK)     // 4096
#define CHUNK   1024             // keys per attention split
#define NSPLIT  (LMAX / CHUNK)   // 4
#define SCALE   0.08838834764831845f     // 128^-0.5
#define EPS     1e-6f

typedef __attribute__((ext_vector_type(2))) float v2f;
typedef __attribute__((ext_vector_type(8))) float v8f;

// =====================================================================
// Kernel A: out[16 x N] = H[16 x K] @ W[N x K]^T  via V_WMMA_F32_16X16X4_F32
// One wave computes one 16x16 tile of the output.  blockDim.x = 256 (8 waves).
// A-frag: lane L holds H[L&15][k0 + 2*(L>>4) + v]          (16x4, 2 VGPRs)
// B-frag: lane L holds W[n0+(L&15)][k0 + 2*(L>>4) + v]     (4x16, 2 VGPRs)
// D     : vgpr v, lane L -> row v + 8*(L>>4), col n0+(L&15)
// =====================================================================
__global__ __launch_bounds__(256) void gemm16_wmma(
    const float* __restrict__ H, const float* __restrict__ W,
    float* __restrict__ out, int K, int N)
{
    const int wave = threadIdx.x >> 5;
    const int lane = threadIdx.x & 31;
    const int n0 = (blockIdx.x * 8 + wave) * 16;
    if (n0 >= N) return;                      // wave-uniform

    const int m     = lane & 15;
    const int khalf = (lane >> 4) << 1;       // 0 or 2

    const float* __restrict__ arow = H + (size_t)m * K + khalf;
    const float* __restrict__ brow = W + (size_t)(n0 + m) * K + khalf;

    v8f c = {};
    for (int k0 = 0; k0 < K; k0 += 16) {
        __builtin_prefetch(brow + k0 + 2048, 0, 0);    // global_prefetch_b8, 8KB ahead
#pragma unroll
        for (int kk = 0; kk < 16; kk += 4) {
            v2f a = *(const v2f*)(arow + k0 + kk);
            v2f b = *(const v2f*)(brow + k0 + kk);
            c = __builtin_amdgcn_wmma_f32_16x16x4_f32(
                    /*neg_a=*/false, a, /*neg_b=*/false, b,
                    /*c_mod=*/(short)0, c, /*reuse_a=*/false, /*reuse_b=*/false);
        }
    }

    const int nn    = n0 + (lane & 15);
    const int mbase = (lane >> 4) << 3;
#pragma unroll
    for (int v = 0; v < 8; ++v)
        out[(size_t)(mbase + v) * N + nn] = c[v];
}

// =====================================================================
// Kernel B: per-head RMSNorm + RoPE, in place inside the qkv workspace.
// grid = (BATCH, NH + NKV), block = 128.
// =====================================================================
__global__ __launch_bounds__(128) void normrope_kernel(
    float* __restrict__ qkv,
    const float* __restrict__ qnw, const float* __restrict__ knw,
    const int* __restrict__ positions,
    const float* __restrict__ cs_cache)
{
    const int b = blockIdx.x;
    const int h = blockIdx.y;
    const int d = threadIdx.x;
    const bool isq = (h < NH);

    float* vec = qkv + (size_t)b * QKV_N + (isq ? h * HD : Q_SIZE + (h - NH) * HD);
    const float* w = isq ? qnw : knw;

    __shared__ __align__(16) float sh[HD];
    __shared__ float red[4];

    float x = vec[d];
    float ss = x * x;
#pragma unroll
    for (int off = 16; off; off >>= 1) ss += __shfl_down(ss, off, 32);
    if ((d & 31) == 0) red[d >> 5] = ss;
    __syncthreads();
    const float tot = red[0] + red[1] + red[2] + red[3];
    const float r = rsqrtf(tot * (1.0f / (float)HD) + EPS);

    sh[d] = x * r * w[d];
    __syncthreads();

    const int pos = positions[b];
    const float* cs = cs_cache + (size_t)pos * HD;
    float o;
    if (d < 64) {
        o = sh[d] * cs[d] - sh[d + 64] * cs[64 + d];
    } else {
        const int j = d - 64;
        o = sh[d] * cs[j] + sh[j] * cs[64 + j];
    }
    vec[d] = o;
}

// =====================================================================
// Kernel C1: flash-decode partial attention over one CHUNK of keys.
// grid = (BATCH, NKV, NSPLIT), block = 256 (8 waves).
// 8 lanes cooperate per key (contiguous 128B per group per step);
// V pass reads each element once, accumulating both GQA heads.
// New-token K/V read from workspace at l == L-1 (no cache mutation).
// =====================================================================
__global__ __launch_bounds__(256) void attn_partial_kernel(
    const float* __restrict__ qkv,
    const float* __restrict__ kcache, const float* __restrict__ vcache,
    const int* __restrict__ block_tables, const int* __restrict__ context_lens,
    float* __restrict__ ws_m, float* __restrict__ ws_s,
    float* __restrict__ ws_acc)
{
    const int b  = blockIdx.x;
    const int kv = blockIdx.y;
    const int sp = blockIdx.z;
    const int tid = threadIdx.x, lane = tid & 31, wid = tid >> 5;

    const int L  = context_lens[b];
    const int l0 = sp * CHUNK;
    const int l1 = min(L, l0 + CHUNK);
    if (l0 >= l1) return;                 // block-uniform

    __shared__ __align__(16) float qs[2][HD];
    __shared__ float sc0[CHUNK], sc1[CHUNK];
    __shared__ float redM[2][8], redS[2][8], bc[4];
    __shared__ float vred[2][2][HD];

    const int* __restrict__ bt = block_tables + (size_t)b * MAXB;
    const float* __restrict__ knew = qkv + (size_t)b * QKV_N + Q_SIZE + kv * HD;
    const float* __restrict__ vnew = knew + KV_SIZE;

    {   // stage q (2 GQA heads) into LDS
        const int g = tid >> 7, d = tid & 127;
        qs[g][d] = qkv[(size_t)b * QKV_N + (kv * 2 + g) * HD + d];
    }
    __syncthreads();

    // ---- pass 1: scores (8-lane cooperative dot), running max --------
    const int dlane = lane & 7;           // dim-group lane 0..7
    const int ksub  = lane >> 3;          // key sub-slot 0..3 within wave
    float m0 = -INFINITY, m1 = -INFINITY;
    for (int lbase = l0; lbase < l1; lbase += 32) {
        const int l = lbase + wid * 4 + ksub;   // NB: wid*4 spans 32 keys over 8 waves
        if (l < l1) {
            const float* kp = (l == L - 1) ? knew
                : kcache + (((size_t)bt[l >> 4] * BLK + (l & 15)) * NKV + kv) * HD;
            float p0 = 0.f, p1 = 0.f;
#pragma unroll
            for (int i = 0; i < 4; ++i) {
                const int off = i * 32 + dlane * 4;
                const float4 kk = *(const float4*)(kp + off);
                const float4 q0 = *(const float4*)(qs[0] + off);
                const float4 q1 = *(const float4*)(qs[1] + off);
                p0 += kk.x * q0.x + kk.y * q0.y + kk.z * q0.z + kk.w * q0.w;
                p1 += kk.x * q1.x + kk.y * q1.y + kk.z * q1.z + kk.w * q1.w;
            }
#pragma unroll
            for (int off = 1; off < 8; off <<= 1) {
                p0 += __shfl_xor(p0, off, 32);
                p1 += __shfl_xor(p1, off, 32);
            }
            p0 *= SCALE; p1 *= SCALE;
            if (dlane == 0) { sc0[l - l0] = p0; sc1[l - l0] = p1; }
            m0 = fmaxf(m0, p0); m1 = fmaxf(m1, p1);
        }
    }
#pragma unroll
    for (int off = 16; off; off >>= 1) {
        m0 = fmaxf(m0, __shfl_xor(m0, off, 32));
        m1 = fmaxf(m1, __shfl_xor(m1, off, 32));
    }
    if (lane == 0) { redM[0][wid] = m0; redM[1][wid] = m1; }
    __syncthreads();
    if (tid == 0) {
        float a = redM[0][0], c = redM[1][0];
        for (int i = 1; i < 8; ++i) { a = fmaxf(a, redM[0][i]); c = fmaxf(c, redM[1][i]); }
        bc[0] = a; bc[1] = c;
    }
    __syncthreads();
    const float gm0 = bc[0], gm1 = bc[1];

    // ---- pass 2: exp + partial sum ------------------------------------
    const int n = l1 - l0;
    float t0 = 0.f, t1 = 0.f;
    for (int i = tid; i < n; i += 256) {
        const float e0 = __expf(sc0[i] - gm0);
        const float e1 = __expf(sc1[i] - gm1);
        sc0[i] = e0; sc1[i] = e1;
        t0 += e0; t1 += e1;
    }
#pragma unroll
    for (int off = 16; off; off >>= 1) {
        t0 += __shfl_xor(t0, off, 32);
        t1 += __shfl_xor(t1, off, 32);
    }
    if (lane == 0) { redS[0][wid] = t0; redS[1][wid] = t1; }
    __syncthreads();
    if (tid == 0) {
        float a = 0.f, c = 0.f;
        for (int i = 0; i < 8; ++i) { a += redS[0][i]; c += redS[1][i]; }
        bc[2] = a; bc[3] = c;
    }
    __syncthreads();

    // ---- pass 3: partial out, V read once for both heads --------------
    const int half = tid >> 7, d = tid & 127;
    float a0 = 0.f, a1 = 0.f;
    for (int l = l0 + half; l < l1; l += 2) {
        const float vv = (l == L - 1) ? vnew[d]
            : vcache[(((size_t)bt[l >> 4] * BLK + (l & 15)) * NKV + kv) * HD + d];
        a0 += sc0[l - l0] * vv;
        a1 += sc1[l - l0] * vv;
    }
    vred[half][0][d] = a0;
    vred[half][1][d] = a1;
    __syncthreads();

    const int g = tid >> 7;
    const float acc = vred[0][g][d] + vred[1][g][d];
    const size_t base = ((size_t)(b * NKV + kv) * NSPLIT + sp) * 2 + g;
    ws_acc[base * HD + d] = acc;
    if (d == 0) { ws_m[base] = bc[g]; ws_s[base] = bc[2 + g]; }
}

// =====================================================================
// Kernel C2: combine split partials.  grid = (BATCH, NKV), block = 256.
// out = sum_s acc_s * e^{m_s - M} / sum_s sum_s * e^{m_s - M}
// =====================================================================
__global__ __launch_bounds__(256) void attn_combine_kernel(
    const float* __restrict__ ws_m, const float* __restrict__ ws_s,
    const float* __restrict__ ws_acc,
    const int* __restrict__ context_lens,
    float* __restrict__ attn_out)
{
    const int b = blockIdx.x, kv = blockIdx.y;
    const int g = threadIdx.x >> 7, d = threadIdx.x & 127;
    const int L = context_lens[b];
    const int ns = (L + CHUNK - 1) / CHUNK;

    const size_t base = ((size_t)(b * NKV + kv) * NSPLIT) * 2 + g;
    float M = -INFINITY;
    for (int s = 0; s < ns; ++s) M = fmaxf(M, ws_m[base + (size_t)s * 2]);
    float tot = 0.f, acc = 0.f;
    for (int s = 0; s < ns; ++s) {
        const size_t bs = base + (size_t)s * 2;
        const float w = __expf(ws_m[bs] - M);
        tot += ws_s[bs] * w;
        acc += ws_acc[bs * HD + d] * w;
    }
    attn_out[(size_t)b * Q_SIZE + (kv * 2 + g) * HD + d] = acc / tot;
}

// =====================================================================
extern "C" void kernel_launch(void* const* d_in, const int* in_sizes, int n_in,
                              void* d_out, int out_size, void* d_ws, size_t ws_size,
                              hipStream_t stream)
{
    const float* hidden   = (const float*)d_in[0];
    const int*   positions= (const int*)  d_in[1];
    const float* qkv_w    = (const float*)d_in[2];
    const float* q_norm_w = (const float*)d_in[3];
    const float* k_norm_w = (const float*)d_in[4];
    const float* o_w      = (const float*)d_in[5];
    const float* cs_cache = (const float*)d_in[6];
    const float* k_cache  = (const float*)d_in[7];
    const float* v_cache  = (const float*)d_in[8];
    // d_in[9] = slot_mapping (unused: new-token slot == gathered index L-1)
    const int*   block_tables = (const int*)d_in[10];
    const int*   context_lens = (const int*)d_in[11];
    float* out = (float*)d_out;

    float* ws_qkv  = (float*)d_ws;                          // [16,4096]
    float* ws_attn = ws_qkv  + (size_t)BATCH * QKV_N;       // [16,2048]
    float* ws_m    = ws_attn + (size_t)BATCH * Q_SIZE;      // [16,8,4,2]
    float* ws_s    = ws_m    + (size_t)BATCH * NKV * NSPLIT * 2;
    float* ws_acc  = ws_s    + (size_t)BATCH * NKV * NSPLIT * 2;  // [...,128]

    // 1) fused QKV projection (WMMA f32)
    gemm16_wmma<<<dim3(QKV_N / 128), 256, 0, stream>>>(hidden, qkv_w, ws_qkv, HID, QKV_N);

    // 2) per-head RMSNorm + RoPE on q and k, in place
    normrope_kernel<<<dim3(BATCH, NH + NKV), 128, 0, stream>>>(
        ws_qkv, q_norm_w, k_norm_w, positions, cs_cache);

    // 3a) flash-decode partial attention, split-K over key chunks
    attn_partial_kernel<<<dim3(BATCH, NKV, NSPLIT), 256, 0, stream>>>(
        ws_qkv, k_cache, v_cache, block_tables, context_lens, ws_m, ws_s, ws_acc);

    // 3b) combine partials
    attn_combine_kernel<<<dim3(BATCH, NKV), 256, 0, stream>>>(
        ws_m, ws_s, ws_acc, context_lens, ws_attn);

    // 4) output projection into d_out (WMMA f32)
    gemm16_wmma<<<dim3(HID / 128), 256, 0, stream>>>(ws_attn, o_w, out, Q_SIZE, HID);
}